// PixelShuffle_23210003267942
// MI455X (gfx1250) — compile-verified
//
#include <hip/hip_runtime.h>
#include <hip/hip_bf16.h>

// Depth-to-space (pixel shuffle), feature-major grouping, R=2.
//   in : [B=8, X=256, Y=256, C=256] fp32
//   out: [B=8, 512, 512, F=64]      fp32
//   out[p*65536 + i*32768 + y*128 + j*64 + f] = in[p*65536 + y*256 + 4f+2i+j]
// Pure permutation: 1.07 GB HBM traffic, 0 FLOPs -> bandwidth roofline ~46us
// at 23.3 TB/s. Strategy: NT b128 loads -> conflict-free LDS staging in exact
// output order (compiler emits ds_store_2addr_b64) -> CDNA5 async 128-bit
// NT stores from LDS (ASYNCcnt path), so waves only generate addresses.

typedef __attribute__((ext_vector_type(4))) float v4f;
typedef __attribute__((ext_vector_type(2))) float v2f;

__global__ __launch_bounds__(256) void pixel_shuffle_f64r2_kernel(
    const float* __restrict__ in, float* __restrict__ out) {
  // LDS staging tile in output order: float index l = i*1024 + yy*128 + j*64 + f
  // (i: output-row parity, yy: 0..7 local y, j: output-col parity, f: feature)
  __shared__ __align__(16) v2f lds2[1024];  // 8 KB

  const unsigned tid = threadIdx.x;
  const unsigned bid = blockIdx.x;
  const unsigned p   = bid >> 5;              // 0..2047  (b*X + x)
  const unsigned y0  = (bid & 31u) * 8u;      // 0..248   (y tile base)

  // ---------------- phase 1: global -> regs -> LDS ----------------
  // Thread owns f-pair (2*fp, 2*fp+1) of pixel (p, y0+yy): channels 8fp..8fp+7.
  const unsigned fp = tid & 31u;              // lane within wave
  const unsigned yy = tid >> 5;               // wave index == local y
  const v4f* __restrict__ in4 = (const v4f*)in;
  const size_t g = (size_t)((p * 256u + y0 + yy) * 64u + 2u * fp);
  const v4f a = __builtin_nontemporal_load(&in4[g]);      // ch 8fp+0..3 (f=2fp)
  const v4f b = __builtin_nontemporal_load(&in4[g + 1]);  // ch 8fp+4..7 (f=2fp+1)

  // value(i,j) for f=2fp is a[2i+j]; for f=2fp+1 it is b[2i+j] -> output-adjacent pair.
  // float2 index = i*512 + yy*64 + j*32 + fp  (conflict-free: 32 lanes x b64 = 64 banks)
  const unsigned base2 = yy * 64u;
  { v2f t = {a.x, b.x}; lds2[       base2 +       fp] = t; }  // i=0, j=0
  { v2f t = {a.y, b.y}; lds2[       base2 + 32u + fp] = t; }  // i=0, j=1
  { v2f t = {a.z, b.z}; lds2[512u + base2 +       fp] = t; }  // i=1, j=0
  { v2f t = {a.w, b.w}; lds2[512u + base2 + 32u + fp] = t; }  // i=1, j=1

  __syncthreads();

  // ---------------- phase 2: LDS -> global, async b128 NT stores ----------------
  // LDS is two 4 KB runs, each contiguous in the output:
  //   region i -> out floats [p*65536 + i*32768 + y0*128, +1024)
  // Flat shared-pointer low 32 bits == LDS byte offset (ISA 10.2 aperture rule).
  const unsigned lbase = (unsigned)(uintptr_t)&lds2[0];
#pragma unroll
  for (int it = 0; it < 2; ++it) {
    const unsigned c      = tid + (unsigned)it * 256u;  // 16B chunk id, 0..511
    const unsigned i      = c >> 8;
    const unsigned cc     = c & 255u;
    const unsigned srcLds = lbase + c * 16u;
    const unsigned long long ofl =
        (unsigned long long)p * 65536ull + (unsigned long long)i * 32768ull +
        (unsigned long long)y0 * 128ull + (unsigned long long)cc * 4ull;
    const unsigned long long gaddr = (unsigned long long)(uintptr_t)out + ofl * 4ull;
    asm volatile("global_store_async_from_lds_b128 %0, %1, off th:TH_STORE_NT"
                 :
                 : "v"(gaddr), "v"(srcLds)
                 : "memory");
  }
  // Drain our async stores (ENDPGM also implies wait-idle; explicit for clarity).
  asm volatile("s_wait_asynccnt 0x0" ::: "memory");
}

extern "C" void kernel_launch(void* const* d_in, const int* in_sizes, int n_in,
                              void* d_out, int out_size, void* d_ws, size_t ws_size,
                              hipStream_t stream) {
  (void)in_sizes; (void)n_in; (void)d_ws; (void)ws_size; (void)out_size;
  const float* in = (const float*)d_in[0];
  float* out = (float*)d_out;
  // 2048 p-rows * 32 y-tiles = 65536 blocks of 256 threads
  dim3 grid(65536), block(256);
  hipLaunchKernelGGL(pixel_shuffle_f64r2_kernel, grid, block, 0, stream, in, out);
}